// DiffusionFlowEmbedder_62139586839116
// MI455X (gfx1250) — compile-verified
//
#include <hip/hip_runtime.h>
#include <hip/hip_bf16.h>
#include <math.h>

typedef float v2f __attribute__((ext_vector_type(2)));
typedef float v8f __attribute__((ext_vector_type(8)));

#define NN 4096
#define DD 100
#define SIGMA_INV 2.0f   // 1/0.5

// ---------------------------------------------------------------------------
// WMMA fp32 GEMM: C = A * B, all NN x NN row-major fp32.
// Block tile 128x128, 8 waves (4Mx2N), wave tile 32x64 = 2x4 WMMA 16x16 tiles.
// BK=32, double-buffered LDS, B staged transposed so every fragment is one
// aligned ds_load_b64.
// ---------------------------------------------------------------------------
#define BM 128
#define BN 128
#define BK 32

__global__ __launch_bounds__(256)
void gemm4096_wmma(float* __restrict__ C, const float* __restrict__ A,
                   const float* __restrict__ B) {
  __shared__ float As[2][BM][BK + 4];   // stride 36 dwords: 16B-aligned rows, conflict-free
  __shared__ float Bt[2][BN][BK + 2];   // transposed B tile, stride 34 dwords (even -> b64 ok)
  const int tid  = threadIdx.x;
  const int lane = tid & 31;
  const int wave = tid >> 5;            // 0..7
  const int m0   = (wave >> 1) * 32;    // wave M origin (4 waves down)
  const int n0   = (wave & 1) * 64;     // wave N origin (2 waves across)
  const int blockM = blockIdx.y * BM;
  const int blockN = blockIdx.x * BN;
  const int hi = lane >> 4;             // lane half: K pair select / M+8
  const int lo = lane & 15;

  const float4* Ag = (const float4*)(A + (size_t)blockM * NN);
  const float4* Bg = (const float4*)(B + blockN);

  v8f acc[2][4] = {};
  float4 ra[4], rb[4];

  auto loadTiles = [&](int kb) {
    #pragma unroll
    for (int p = 0; p < 4; ++p) {        // A: 128x32 = 1024 float4
      int q = p * 256 + tid;
      int r = q >> 3, c4 = q & 7;        // 8 float4 per A row
      ra[p] = Ag[(size_t)r * (NN / 4) + (kb >> 2) + c4];
    }
    #pragma unroll
    for (int p = 0; p < 4; ++p) {        // B: 32x128 = 1024 float4
      int q = p * 256 + tid;
      int r = q >> 5, c4 = q & 31;       // 32 float4 per B row
      rb[p] = Bg[(size_t)(kb + r) * (NN / 4) + c4];
    }
  };
  auto stageTiles = [&](int nbuf) {
    #pragma unroll
    for (int p = 0; p < 4; ++p) {
      int q = p * 256 + tid;
      int r = q >> 3, c4 = q & 7;
      *(float4*)&As[nbuf][r][c4 * 4] = ra[p];
    }
    #pragma unroll
    for (int p = 0; p < 4; ++p) {        // transpose into Bt[n][k]
      int q = p * 256 + tid;
      int r = q >> 5, c4 = q & 31;
      Bt[nbuf][c4 * 4 + 0][r] = rb[p].x;
      Bt[nbuf][c4 * 4 + 1][r] = rb[p].y;
      Bt[nbuf][c4 * 4 + 2][r] = rb[p].z;
      Bt[nbuf][c4 * 4 + 3][r] = rb[p].w;
    }
  };

  loadTiles(0);
  stageTiles(0);
  __syncthreads();

  int buf = 0;
  for (int kb = 0; kb < NN; kb += BK) {
    const bool more = (kb + BK) < NN;
    if (more) {
      loadTiles(kb + BK);               // global loads in flight under the WMMAs
      __builtin_prefetch(A + (size_t)blockM * NN + kb + 2 * BK, 0, 1);
      __builtin_prefetch(B + (size_t)(kb + 2 * BK) * NN + blockN, 0, 1);
    }
    #pragma unroll
    for (int kk = 0; kk < BK; kk += 4) {
      const int kf = kk + hi * 2;       // fragment VGPR v holds K = 2*(lane/16)+v
      v2f a0 = *(const v2f*)&As[buf][m0 + lo][kf];
      v2f a1 = *(const v2f*)&As[buf][m0 + 16 + lo][kf];
      v2f b0 = *(const v2f*)&Bt[buf][n0 + lo][kf];
      v2f b1 = *(const v2f*)&Bt[buf][n0 + 16 + lo][kf];
      v2f b2 = *(const v2f*)&Bt[buf][n0 + 32 + lo][kf];
      v2f b3 = *(const v2f*)&Bt[buf][n0 + 48 + lo][kf];
      acc[0][0] = __builtin_amdgcn_wmma_f32_16x16x4_f32(false, a0, false, b0, (short)0, acc[0][0], false, false);
      acc[0][1] = __builtin_amdgcn_wmma_f32_16x16x4_f32(false, a0, false, b1, (short)0, acc[0][1], false, false);
      acc[0][2] = __builtin_amdgcn_wmma_f32_16x16x4_f32(false, a0, false, b2, (short)0, acc[0][2], false, false);
      acc[0][3] = __builtin_amdgcn_wmma_f32_16x16x4_f32(false, a0, false, b3, (short)0, acc[0][3], false, false);
      acc[1][0] = __builtin_amdgcn_wmma_f32_16x16x4_f32(false, a1, false, b0, (short)0, acc[1][0], false, false);
      acc[1][1] = __builtin_amdgcn_wmma_f32_16x16x4_f32(false, a1, false, b1, (short)0, acc[1][1], false, false);
      acc[1][2] = __builtin_amdgcn_wmma_f32_16x16x4_f32(false, a1, false, b2, (short)0, acc[1][2], false, false);
      acc[1][3] = __builtin_amdgcn_wmma_f32_16x16x4_f32(false, a1, false, b3, (short)0, acc[1][3], false, false);
    }
    if (more) {
      stageTiles(buf ^ 1);
      __syncthreads();
      buf ^= 1;
    }
  }

  // epilogue: C/D layout -> VGPR r holds (M = r + 8*(lane/16), N = lane%16)
  #pragma unroll
  for (int i = 0; i < 2; ++i) {
    #pragma unroll
    for (int j = 0; j < 4; ++j) {
      const int gm = blockM + m0 + i * 16 + hi * 8;
      const int gn = blockN + n0 + j * 16 + lo;
      #pragma unroll
      for (int r = 0; r < 8; ++r)
        C[(size_t)(gm + r) * NN + gn] = acc[i][j][r];
    }
  }
}

// ---------------------------------------------------------------------------
// Graph affinity: a_i = 2*(||f_i||^2 + f_i.x_i);  P[i,j] = exp(2*f_i.x_j - a_i)
// ---------------------------------------------------------------------------
__global__ void graph_row_scalars(const float* __restrict__ X,
                                  const float* __restrict__ F,
                                  float* __restrict__ arow) {
  int i = blockIdx.x * blockDim.x + threadIdx.x;
  if (i >= NN) return;
  float ff = 0.f, fx = 0.f;
  for (int k = 0; k < DD; ++k) {
    float f = F[i * DD + k];
    ff += f * f;
    fx += f * X[i * DD + k];
  }
  arow[i] = (ff + fx) * SIGMA_INV;
}

__global__ __launch_bounds__(256)
void affinity_graph(const float* __restrict__ X, const float* __restrict__ F,
                    const float* __restrict__ arow, float* __restrict__ P) {
  __shared__ float Fi[16][DD + 1];
  __shared__ float Xj[16][DD + 1];
  const int bi = blockIdx.y * 16, bj = blockIdx.x * 16;
  for (int q = threadIdx.x; q < 16 * DD; q += 256) {
    int r = q / DD, c = q % DD;
    Fi[r][c] = F[(bi + r) * DD + c];
    Xj[r][c] = X[(bj + r) * DD + c];
  }
  __syncthreads();
  const int il = threadIdx.x >> 4, jl = threadIdx.x & 15;
  float dot = 0.f;
  for (int k = 0; k < DD; ++k) dot += Fi[il][k] * Xj[jl][k];
  P[(size_t)(bi + il) * NN + bj + jl] = expf(dot * SIGMA_INV - arow[bi + il]);
}

// ---------------------------------------------------------------------------
// Encoder MLP: X(4096x100) -> relu(100) -> relu(10) -> emb(2)
// ---------------------------------------------------------------------------
__global__ void encoder_mlp(const float* __restrict__ X,
                            const float* __restrict__ W1, const float* __restrict__ b1,
                            const float* __restrict__ W2, const float* __restrict__ b2,
                            const float* __restrict__ W3, const float* __restrict__ b3,
                            float* __restrict__ emb) {
  int i = blockIdx.x * blockDim.x + threadIdx.x;
  if (i >= NN) return;
  float xr[DD];
  for (int k = 0; k < DD; ++k) xr[k] = X[i * DD + k];
  float h1[100];
  for (int o = 0; o < 100; ++o) {
    float s = b1[o];
    for (int k = 0; k < DD; ++k) s += xr[k] * W1[k * 100 + o];
    h1[o] = fmaxf(s, 0.f);
  }
  float h2[10];
  for (int o = 0; o < 10; ++o) {
    float s = b2[o];
    for (int k = 0; k < 100; ++k) s += h1[k] * W2[k * 10 + o];
    h2[o] = fmaxf(s, 0.f);
  }
  for (int o = 0; o < 2; ++o) {
    float s = b3[o];
    for (int k = 0; k < 10; ++k) s += h2[k] * W3[k * 2 + o];
    emb[i * 2 + o] = s;
  }
}

// ---------------------------------------------------------------------------
// Flow-artist MLP (tanh): emb(2)->10->10->ef(2); also b_i = 2*(||ef||^2 + ef.emb_i)
// ---------------------------------------------------------------------------
__global__ void fa_and_scalars(const float* __restrict__ emb,
                               const float* __restrict__ W1, const float* __restrict__ b1,
                               const float* __restrict__ W2, const float* __restrict__ b2,
                               const float* __restrict__ W3, const float* __restrict__ b3,
                               float* __restrict__ ef, float* __restrict__ brow) {
  int i = blockIdx.x * blockDim.x + threadIdx.x;
  if (i >= NN) return;
  float x0 = emb[i * 2 + 0], x1 = emb[i * 2 + 1];
  float h1[10], h2[10];
  for (int o = 0; o < 10; ++o)
    h1[o] = tanhf(b1[o] + x0 * W1[0 * 10 + o] + x1 * W1[1 * 10 + o]);
  for (int o = 0; o < 10; ++o) {
    float s = b2[o];
    for (int k = 0; k < 10; ++k) s += h1[k] * W2[k * 10 + o];
    h2[o] = tanhf(s);
  }
  float f0 = b3[0], f1 = b3[1];
  for (int k = 0; k < 10; ++k) { f0 += h2[k] * W3[k * 2 + 0]; f1 += h2[k] * W3[k * 2 + 1]; }
  ef[i * 2 + 0] = f0; ef[i * 2 + 1] = f1;
  brow[i] = (f0 * f0 + f1 * f1 + f0 * x0 + f1 * x1) * SIGMA_INV;
}

// A[i,j] = exp(2*ef_i.emb_j - b_i)   (e = 2)
__global__ void affinity_emb(const float* __restrict__ emb, const float* __restrict__ ef,
                             const float* __restrict__ brow, float* __restrict__ A) {
  int j = blockIdx.x * blockDim.x + threadIdx.x;
  int i = blockIdx.y;
  float d = ef[i * 2 + 0] * emb[j * 2 + 0] + ef[i * 2 + 1] * emb[j * 2 + 1];
  A[(size_t)i * NN + j] = expf(d * SIGMA_INV - brow[i]);
}

__global__ __launch_bounds__(256)
void row_sums(const float* __restrict__ A, float* __restrict__ rs) {
  __shared__ float red[256];
  const int i = blockIdx.x;
  float s = 0.f;
  for (int j = threadIdx.x; j < NN; j += 256) s += A[(size_t)i * NN + j];
  red[threadIdx.x] = s;
  __syncthreads();
  for (int w = 128; w > 0; w >>= 1) {
    if (threadIdx.x < w) red[threadIdx.x] += red[threadIdx.x + w];
    __syncthreads();
  }
  if (threadIdx.x == 0) rs[i] = 1.0f / red[0];
}

__global__ void row_normalize(float* __restrict__ A, const float* __restrict__ rs) {
  int j = blockIdx.x * blockDim.x + threadIdx.x;
  int i = blockIdx.y;
  A[(size_t)i * NN + j] *= rs[i];
}

// ---------------------------------------------------------------------------
// Decoder MLP + reconstruction SSE:  emb(2)->relu(10)->relu(100)->recon(100)
// ---------------------------------------------------------------------------
__global__ void recon_loss(const float* __restrict__ emb, const float* __restrict__ X,
                           const float* __restrict__ W1, const float* __restrict__ b1,
                           const float* __restrict__ W2, const float* __restrict__ b2,
                           const float* __restrict__ W3, const float* __restrict__ b3,
                           float* __restrict__ acc) {
  int i = blockIdx.x * blockDim.x + threadIdx.x;
  if (i >= NN) return;
  float x0 = emb[i * 2 + 0], x1 = emb[i * 2 + 1];
  float h1[10], h2[100];
  for (int o = 0; o < 10; ++o)
    h1[o] = fmaxf(b1[o] + x0 * W1[0 * 10 + o] + x1 * W1[1 * 10 + o], 0.f);
  for (int o = 0; o < 100; ++o) {
    float s = b2[o];
    for (int k = 0; k < 10; ++k) s += h1[k] * W2[k * 100 + o];
    h2[o] = fmaxf(s, 0.f);
  }
  float sse = 0.f;
  for (int o = 0; o < DD; ++o) {
    float s = b3[o];
    for (int k = 0; k < 100; ++k) s += h2[k] * W3[k * DD + o];
    float d = s - X[i * DD + o];
    sse += d * d;
  }
  atomicAdd(acc, sse);
}

// sum Pgt * (log Pgt - log Pet)
__global__ __launch_bounds__(256)
void kl_sum(const float* __restrict__ Pgt, const float* __restrict__ Pet,
            float* __restrict__ acc) {
  __shared__ float red[256];
  size_t stride = (size_t)gridDim.x * blockDim.x;
  float s = 0.f;
  for (size_t idx = (size_t)blockIdx.x * blockDim.x + threadIdx.x;
       idx < (size_t)NN * NN; idx += stride) {
    float g = Pgt[idx];
    s += g * (logf(g) - logf(Pet[idx]));
  }
  red[threadIdx.x] = s;
  __syncthreads();
  for (int w = 128; w > 0; w >>= 1) {
    if (threadIdx.x < w) red[threadIdx.x] += red[threadIdx.x + w];
    __syncthreads();
  }
  if (threadIdx.x == 0) atomicAdd(acc, red[0]);
}

__global__ void zero_acc(float* acc) {
  if (threadIdx.x < 2) acc[threadIdx.x] = 0.f;
}

__global__ void finalize(const float* __restrict__ acc, float* __restrict__ out) {
  out[0] = acc[1] / (float)NN + acc[0] / (float)((size_t)NN * DD);
}

// ---------------------------------------------------------------------------
extern "C" void kernel_launch(void* const* d_in, const int* in_sizes, int n_in,
                              void* d_out, int out_size, void* d_ws, size_t ws_size,
                              hipStream_t stream) {
  (void)in_sizes; (void)n_in; (void)out_size; (void)ws_size;
  const float* X  = (const float*)d_in[0];
  const float* Fl = (const float*)d_in[1];
  const float* eW1 = (const float*)d_in[2];  const float* eb1 = (const float*)d_in[3];
  const float* eW2 = (const float*)d_in[4];  const float* eb2 = (const float*)d_in[5];
  const float* eW3 = (const float*)d_in[6];  const float* eb3 = (const float*)d_in[7];
  const float* dW1 = (const float*)d_in[8];  const float* db1 = (const float*)d_in[9];
  const float* dW2 = (const float*)d_in[10]; const float* db2 = (const float*)d_in[11];
  const float* dW3 = (const float*)d_in[12]; const float* db3 = (const float*)d_in[13];
  const float* fW1 = (const float*)d_in[14]; const float* fb1 = (const float*)d_in[15];
  const float* fW2 = (const float*)d_in[16]; const float* fb2 = (const float*)d_in[17];
  const float* fW3 = (const float*)d_in[18]; const float* fb3 = (const float*)d_in[19];
  float* out = (float*)d_out;

  const size_t MAT = (size_t)NN * NN;          // elements per NxN matrix
  char* ws = (char*)d_ws;
  float* P0 = (float*)ws;                      // 64 MB
  float* P1 = (float*)(ws + MAT * sizeof(float));
  float* P2 = (float*)(ws + 2 * MAT * sizeof(float));
  float* arow   = (float*)(ws + 3 * MAT * sizeof(float));
  float* brow   = arow + NN;
  float* emb    = brow + NN;                   // 4096 x 2
  float* ef     = emb + 2 * NN;                // 4096 x 2
  float* rsums  = ef + 2 * NN;
  float* acc    = rsums + NN;                  // [0]=recon SSE, [1]=KL sum

  dim3 gemmGrid(NN / BN, NN / BM), gemmBlk(256);
  dim3 affGrid(NN / 16, NN / 16), affBlk(256);
  dim3 rowGrid(NN / 256), rowBlk(256);
  dim3 ewGrid(NN / 256, NN), ewBlk(256);

  // graph side: P0 = P_graph ; P1 = P^2 ; P2 = P^4 (kept)
  graph_row_scalars<<<rowGrid, rowBlk, 0, stream>>>(X, Fl, arow);
  affinity_graph<<<affGrid, affBlk, 0, stream>>>(X, Fl, arow, P0);
  gemm4096_wmma<<<gemmGrid, gemmBlk, 0, stream>>>(P1, P0, P0);
  gemm4096_wmma<<<gemmGrid, gemmBlk, 0, stream>>>(P2, P1, P1);

  // embedding side
  encoder_mlp<<<rowGrid, rowBlk, 0, stream>>>(X, eW1, eb1, eW2, eb2, eW3, eb3, emb);
  fa_and_scalars<<<rowGrid, rowBlk, 0, stream>>>(emb, fW1, fb1, fW2, fb2, fW3, fb3, ef, brow);
  affinity_emb<<<ewGrid, ewBlk, 0, stream>>>(emb, ef, brow, P0);   // P0 = A
  row_sums<<<dim3(NN), dim3(256), 0, stream>>>(P0, rsums);
  row_normalize<<<ewGrid, ewBlk, 0, stream>>>(P0, rsums);          // P0 = P_emb
  gemm4096_wmma<<<gemmGrid, gemmBlk, 0, stream>>>(P1, P0, P0);     // P1 = Pe^2
  gemm4096_wmma<<<gemmGrid, gemmBlk, 0, stream>>>(P0, P1, P1);     // P0 = Pe^4

  // losses
  zero_acc<<<dim3(1), dim3(32), 0, stream>>>(acc);
  recon_loss<<<rowGrid, rowBlk, 0, stream>>>(emb, X, dW1, db1, dW2, db2, dW3, db3, acc + 0);
  kl_sum<<<dim3(4096), dim3(256), 0, stream>>>(P2, P0, acc + 1);
  finalize<<<dim3(1), dim3(1), 0, stream>>>(acc, out);
}